// Node_encoder_36086315221303
// MI455X (gfx1250) — compile-verified
//
#include <hip/hip_runtime.h>

#define BB 4
#define NODES 50
#define NFD 16
#define LSEQ 64
#define HID 64
#define DSTATE 128
#define DTRANK 4
#define OUT_DIM 32
#define OUT_LEN 32
#define LN_EPS 1e-5f

// flat output offsets: z_x | x_mu | x_sigma, each (4,32,50,32)
#define OUT_MU  204800
#define OUT_SG  409600

typedef float v2f __attribute__((ext_vector_type(2)));
typedef float v8f __attribute__((ext_vector_type(8)));

__device__ __forceinline__ float fsilu(float x) { return x / (1.f + __expf(-x)); }
__device__ __forceinline__ float fsoftplus(float x) {
  return x > 20.f ? x : __logf(1.f + __expf(x));
}

// C[(mtiles*16) x N] = A_lds[row0 .. row0+mtiles*16) x K] @ W^T  (+bias, +act)
// W is row-major (N_out x K). act: 0=none, 1=silu, 2=softplus.
// Uses V_WMMA_F32_16X16X4_F32; all control flow is wave-uniform.
__device__ void wmma_gemm(const float* Alds, int lda, int row0, int mtiles,
                          const float* __restrict__ W, int ldw, int N, int K,
                          const float* __restrict__ bias, int act,
                          float* Clds, int ldc, int tid)
{
  const int wave = tid >> 5;
  const int lane = tid & 31;
  const int lm   = lane & 15;   // A row-in-tile / B col-in-tile
  const int hi   = lane >> 4;   // selects K pair {0,1} vs {2,3}
  const int ntiles = N >> 4;
  const int tiles  = mtiles * ntiles;
  for (int t = wave; t < tiles; t += 8) {
    const int mi = t / ntiles, ni = t - mi * ntiles;
    v8f c = {0.f,0.f,0.f,0.f,0.f,0.f,0.f,0.f};
    const float* arow = Alds + (row0 + mi * 16 + lm) * lda;
    const float* wrow = W + (ni * 16 + lm) * ldw;
    for (int k0 = 0; k0 < K; k0 += 4) {
      const int ka = k0 + hi * 2;
      v2f av, bv;
      av.x = arow[ka];     av.y = arow[ka + 1];
      bv.x = wrow[ka];     bv.y = wrow[ka + 1];
      c = __builtin_amdgcn_wmma_f32_16x16x4_f32(false, av, false, bv,
                                                (short)0, c, false, false);
    }
    const int col = ni * 16 + lm;
    const float bb = bias ? bias[col] : 0.f;
    #pragma unroll
    for (int r = 0; r < 8; ++r) {
      float v = c[r] + bb;
      if (act == 1) v = fsilu(v);
      else if (act == 2) v = fsoftplus(v);
      Clds[(mi * 16 + r + hi * 8) * ldc + col] = v;
    }
  }
}

__global__ __launch_bounds__(256)
void node_encoder_kernel(
    const float* __restrict__ x, const float* __restrict__ nodes_mask,
    const float* __restrict__ emb_w, const float* __restrict__ emb_b,
    const float* __restrict__ dim_layer_w, const float* __restrict__ dim_layer_b,
    const float* __restrict__ dim_out_w, const float* __restrict__ dim_out_b,
    const float* __restrict__ in_proj_w,
    const float* __restrict__ conv_w, const float* __restrict__ conv_b,
    const float* __restrict__ x_proj_w,
    const float* __restrict__ dt_proj_w, const float* __restrict__ dt_proj_b,
    const float* __restrict__ A_log, const float* __restrict__ Dp,
    const float* __restrict__ out_proj_w,
    const float* __restrict__ ln_w, const float* __restrict__ ln_b,
    const float* __restrict__ mu_w, const float* __restrict__ mu_b,
    const float* __restrict__ sigma_w, const float* __restrict__ sigma_b,
    float* __restrict__ out)
{
  __shared__ float sA[64 * 64];   // E -> xi -> dt -> silu(out_proj)
  __shared__ float sB[64 * 64];   // H1 -> xc -> dim_out/LN result
  __shared__ float sZ[64 * 64];   // z  -> y (scan output, gated)
  __shared__ float sC[4096];      // x-staging / dt_in / scan B,C chunks / LN scratch

  const int tid = threadIdx.x;
  const int seq = blockIdx.x;       // s = node*B + b
  const int nnode = seq / BB;
  const int b = seq - nnode * BB;

  // ---- stage x[b,n,:,:] (NFD x L) transposed into sC as (L x 16) ----
  const float* xs = x + (size_t)(b * NODES + nnode) * NFD * LSEQ;
  for (int i = tid; i < NFD * LSEQ; i += 256) {
    int c = i >> 6, l = i & 63;
    sC[l * 16 + c] = xs[c * 64 + l];
  }
  __syncthreads();
  // ---- E = Xt @ emb_w^T + emb_b -> sA ----
  wmma_gemm(sC, 16, 0, 4, emb_w, NFD, HID, NFD, emb_b, 0, sA, 64, tid);
  __syncthreads();
  // ---- H1 = E @ dim_layer_w^T + b -> sB ----
  wmma_gemm(sA, 64, 0, 4, dim_layer_w, 64, 64, 64, dim_layer_b, 0, sB, 64, tid);
  __syncthreads();
  // ---- in_proj: xi -> sA, z -> sZ ----
  wmma_gemm(sB, 64, 0, 4, in_proj_w,           64, 64, 64, nullptr, 0, sA, 64, tid);
  wmma_gemm(sB, 64, 0, 4, in_proj_w + 64 * 64, 64, 64, 64, nullptr, 0, sZ, 64, tid);
  __syncthreads();
  // ---- depthwise conv (k=2) + silu -> xc in sB ----
  for (int i = tid; i < 4096; i += 256) {
    int l = i >> 6, d = i & 63;
    float prev = (l > 0) ? sA[(l - 1) * 64 + d] : 0.f;
    float v = prev * conv_w[d * 2] + sA[i] * conv_w[d * 2 + 1] + conv_b[d];
    sB[i] = fsilu(v);
  }
  __syncthreads();
  // ---- dt_in = xc @ x_proj_w[0:4]^T -> sC (64x16, cols 0..3 valid) ----
  wmma_gemm(sB, 64, 0, 4, x_proj_w, 64, 16, 64, nullptr, 0, sC, 16, tid);
  __syncthreads();
  // ---- dt = softplus(dt_in @ dt_proj_w^T + b) -> sA ----
  wmma_gemm(sC, 16, 0, 4, dt_proj_w, DTRANK, 64, 4, dt_proj_b, 2, sA, 64, tid);
  __syncthreads();

  // ---- selective scan: thread <-> (d = tid/4, 32-wide n slice) ----
  const int d  = tid >> 2;
  const int j  = tid & 3;
  const int n0 = j * 32;
  float Ar[32], hst[32];
  #pragma unroll
  for (int i = 0; i < 32; ++i) {
    Ar[i]  = -__expf(A_log[d * DSTATE + n0 + i]);
    hst[i] = 0.f;
  }
  const float Dd = Dp[d];

  for (int ch = 0; ch < 4; ++ch) {
    const int l0 = ch * 16;
    // B,C rows for this 16-step chunk via WMMA -> sC[0..2047], sC[2048..4095]
    wmma_gemm(sB, 64, l0, 1, x_proj_w + DTRANK * 64,            64, DSTATE, 64,
              nullptr, 0, sC,        DSTATE, tid);
    wmma_gemm(sB, 64, l0, 1, x_proj_w + (DTRANK + DSTATE) * 64, 64, DSTATE, 64,
              nullptr, 0, sC + 2048, DSTATE, tid);
    __syncthreads();
    for (int lr = 0; lr < 16; ++lr) {
      const int l = l0 + lr;
      const float dt_l = sA[l * 64 + d];
      const float u_l  = sB[l * 64 + d];
      const float du   = dt_l * u_l;
      float acc = 0.f;
      #pragma unroll
      for (int i = 0; i < 32; ++i) {
        float bm = sC[lr * DSTATE + n0 + i];
        float cm = sC[2048 + lr * DSTATE + n0 + i];
        hst[i] = __expf(dt_l * Ar[i]) * hst[i] + du * bm;
        acc += hst[i] * cm;
      }
      acc += __shfl_xor(acc, 1);
      acc += __shfl_xor(acc, 2);
      if (j == 0) {
        float y  = acc + u_l * Dd;
        float zv = sZ[l * 64 + d];
        sZ[l * 64 + d] = y * fsilu(zv);   // gate; same-thread read-then-write
      }
    }
    __syncthreads();
  }

  // ---- silu(y @ out_proj^T) -> sA ----
  wmma_gemm(sZ, 64, 0, 4, out_proj_w, 64, 64, 64, nullptr, 1, sA, 64, tid);
  __syncthreads();
  // ---- H2 = sA @ dim_out^T + b -> sB ----
  wmma_gemm(sA, 64, 0, 4, dim_out_w, 64, 64, 64, dim_out_b, 0, sB, 64, tid);
  __syncthreads();

  // ---- per-sequence layernorm over all 4096 elements of sB ----
  float sum = 0.f, sq = 0.f;
  for (int k = 0; k < 16; ++k) {
    float v = sB[tid + k * 256];
    sum += v; sq += v * v;
  }
  #pragma unroll
  for (int o = 16; o > 0; o >>= 1) {
    sum += __shfl_xor(sum, o);
    sq  += __shfl_xor(sq, o);
  }
  if ((tid & 31) == 0) { sC[tid >> 5] = sum; sC[8 + (tid >> 5)] = sq; }
  __syncthreads();
  if (tid == 0) {
    float S = 0.f, Q = 0.f;
    for (int w = 0; w < 8; ++w) { S += sC[w]; Q += sC[8 + w]; }
    float mean = S * (1.f / 4096.f);
    float var  = Q * (1.f / 4096.f) - mean * mean;
    sC[16] = mean;
    sC[17] = rsqrtf(var + LN_EPS);
  }
  __syncthreads();
  {
    const float mean = sC[16], rstd = sC[17];
    for (int k = 0; k < 16; ++k) {
      int idx = tid + k * 256;
      int l = idx >> 6, hc = idx & 63;
      sB[idx] = (sB[idx] - mean) * rstd * ln_w[l * 64 + hc] + ln_b[l * 64 + hc];
    }
  }
  __syncthreads();

  // ---- mu/sigma heads on last 32 tokens: one 16x16 tile per wave ----
  {
    const float mask = nodes_mask[b * NODES + nnode];
    const int wave = tid >> 5, lane = tid & 31, lm = lane & 15, hi = lane >> 4;
    const int sel = wave >> 2;            // 0 = mu, 1 = sigma
    const int tt = wave & 3, mi = tt >> 1, ni = tt & 1;
    const float* W    = sel ? sigma_w : mu_w;
    const float* bias = sel ? sigma_b : mu_b;
    const float* arow = sB + (32 + mi * 16 + lm) * 64;   // tokens 32..63
    const float* wrow = W + (ni * 16 + lm) * 64;
    v8f c = {0.f,0.f,0.f,0.f,0.f,0.f,0.f,0.f};
    for (int k0 = 0; k0 < 64; k0 += 4) {
      const int ka = k0 + hi * 2;
      v2f av, bv;
      av.x = arow[ka]; av.y = arow[ka + 1];
      bv.x = wrow[ka]; bv.y = wrow[ka + 1];
      c = __builtin_amdgcn_wmma_f32_16x16x4_f32(false, av, false, bv,
                                                (short)0, c, false, false);
    }
    const int o = ni * 16 + lm;
    const float bo = bias[o];
    #pragma unroll
    for (int r = 0; r < 8; ++r) {
      const int lp = mi * 16 + r + hi * 8;         // 0..31
      float v = fsilu(c[r] + bo);
      size_t idx = ((size_t)(b * OUT_DIM + o) * NODES + nnode) * OUT_LEN + lp;
      if (sel == 0) {
        out[OUT_MU + idx] = v;        // x_mu
        out[idx]          = v * mask; // z_x
      } else {
        out[OUT_SG + idx] = v;        // x_sigma
      }
    }
  }
}

extern "C" void kernel_launch(void* const* d_in, const int* in_sizes, int n_in,
                              void* d_out, int out_size, void* d_ws, size_t ws_size,
                              hipStream_t stream) {
  (void)in_sizes; (void)n_in; (void)out_size; (void)d_ws; (void)ws_size;
  const float* x           = (const float*)d_in[0];
  const float* nodes_mask  = (const float*)d_in[1];
  // d_in[2] = kl_mask (unused by forward outputs)
  const float* emb_w       = (const float*)d_in[3];
  const float* emb_b       = (const float*)d_in[4];
  const float* dim_layer_w = (const float*)d_in[5];
  const float* dim_layer_b = (const float*)d_in[6];
  const float* dim_out_w   = (const float*)d_in[7];
  const float* dim_out_b   = (const float*)d_in[8];
  const float* in_proj_w   = (const float*)d_in[9];
  const float* conv_w      = (const float*)d_in[10];
  const float* conv_b      = (const float*)d_in[11];
  const float* x_proj_w    = (const float*)d_in[12];
  const float* dt_proj_w   = (const float*)d_in[13];
  const float* dt_proj_b   = (const float*)d_in[14];
  const float* A_log       = (const float*)d_in[15];
  const float* Dp          = (const float*)d_in[16];
  const float* out_proj_w  = (const float*)d_in[17];
  const float* ln_w        = (const float*)d_in[18];
  const float* ln_b        = (const float*)d_in[19];
  const float* mu_w        = (const float*)d_in[20];
  const float* mu_b        = (const float*)d_in[21];
  const float* sigma_w     = (const float*)d_in[22];
  const float* sigma_b     = (const float*)d_in[23];

  node_encoder_kernel<<<NODES * BB, 256, 0, stream>>>(
      x, nodes_mask, emb_w, emb_b, dim_layer_w, dim_layer_b,
      dim_out_w, dim_out_b, in_proj_w, conv_w, conv_b, x_proj_w,
      dt_proj_w, dt_proj_b, A_log, Dp, out_proj_w, ln_w, ln_b,
      mu_w, mu_b, sigma_w, sigma_b, (float*)d_out);
}